// PCAGAT_644245094862
// MI455X (gfx1250) — compile-verified
//
#include <hip/hip_runtime.h>

#define NH 4
#define HD 32
#define IN_DIM 128
#define ATT_K 192
#define GATE_K 64
#define HSZ (1u << 19)
#define HMASK (HSZ - 1u)
#define HEMPTY 0xFFFFFFFFu

typedef float v2f __attribute__((ext_vector_type(2)));
typedef float v8f __attribute__((ext_vector_type(8)));
typedef unsigned int u32x4 __attribute__((ext_vector_type(4)));
typedef int i32x4 __attribute__((ext_vector_type(4)));
typedef int i32x8 __attribute__((ext_vector_type(8)));

__device__ __forceinline__ float leakyf(float x) { return x > 0.f ? x : 0.2f * x; }
__device__ __forceinline__ float sigmoidf(float x) { return 1.f / (1.f + __expf(-x)); }

__device__ __forceinline__ void atomicMaxF(float* addr, float val) {
  int* ia = (int*)addr;
  int old = *ia;
  while (__int_as_float(old) < val) {
    int assumed = old;
    old = atomicCAS(ia, assumed, __float_as_int(val));
    if (old == assumed) break;
  }
}

// ---- Tensor Data Mover: 1-row DMA of n_dwords*4 bytes global -> LDS --------
// D# layout per CDNA5 ISA ch.8: group0 = {count/type/addr}, group1 = dims.
// 6-arg builtin form (clang-23 / therock-10.0 headers).
__device__ __forceinline__ void tdm_load_1d(const void* gptr, void* lptr, unsigned n_dwords) {
  unsigned long long ga = (unsigned long long)gptr;
  unsigned lds = (unsigned)(unsigned long long)lptr;  // LDS aperture: addr[31:0] = LDS byte offset
  u32x4 g0;
  g0.x = 0x1u;                                          // count=1, user mode
  g0.y = lds;                                           // lds_addr
  g0.z = (unsigned)(ga & 0xFFFFFFFFu);                  // global_addr[31:0]
  g0.w = (unsigned)((ga >> 32) & 0x01FFFFFFu) | 0x80000000u;  // addr[56:32] | type=2
  i32x8 g1;
  g1[0] = 0x00020000;                                   // workgroup_mask=0, data_size=4B
  g1[1] = (int)((n_dwords & 0xFFFFu) << 16);            // tensor_dim0[15:0] @ bits63:48
  g1[2] = (int)(((n_dwords >> 16) & 0xFFFFu) | (1u << 16));  // dim0 hi | tensor_dim1=1
  g1[3] = (int)((n_dwords & 0xFFFFu) << 16);            // tensor_dim1 hi=0 | tile_dim0
  g1[4] = 1;                                            // tile_dim1=1, tile_dim2=0
  g1[5] = (int)n_dwords;                                // tensor_dim0_stride[31:0]
  g1[6] = 0;                                            // stride hi, dim1_stride lo
  g1[7] = 0;
  i32x4 z4 = {0, 0, 0, 0};
  i32x8 z8 = {0, 0, 0, 0, 0, 0, 0, 0};
  __builtin_amdgcn_tensor_load_to_lds(g0, g1, z4, z4, z8, 0);
}

// ---------------- init -------------------------------------------------------
__global__ void k_init_mx(float* mx, int n) {
  int i = blockIdx.x * blockDim.x + threadIdx.x;
  if (i < n) mx[i] = -3.0e38f;
}

// ---------------- constraint hash build -------------------------------------
__global__ void k_hash_build(const int* __restrict__ c_src, const int* __restrict__ c_tgt,
                             const float* __restrict__ c_val, const int* __restrict__ c_typ,
                             const float* __restrict__ lambda_c, unsigned* __restrict__ keys,
                             float* __restrict__ vals, float* __restrict__ sum_ws,
                             int n_c, int n_nodes, int n_ctype) {
  int i = blockIdx.x * blockDim.x + threadIdx.x;
  if (i >= n_c) return;
  unsigned key = (unsigned)c_src[i] * (unsigned)n_nodes + (unsigned)c_tgt[i];
  int t = c_typ[i];
  float v = c_val[i];
  float wsv[NH];
#pragma unroll
  for (int h = 0; h < NH; ++h) {
    float lc = sigmoidf(lambda_c[h * n_ctype + t]) * 0.99f + 0.01f;  // *(1-min)+min
    wsv[h] = lc * v;
    atomicAdd(&sum_ws[h], wsv[h]);
  }
  unsigned slot = (key * 2654435761u) & HMASK;
  while (true) {
    unsigned prev = atomicCAS(&keys[slot], HEMPTY, key);
    if (prev == HEMPTY || prev == key) {
#pragma unroll
      for (int h = 0; h < NH; ++h) atomicAdd(&vals[(size_t)h * HSZ + slot], wsv[h]);
      break;
    }
    slot = (slot + 1) & HMASK;
  }
}

// ---------------- Wh = h @ [W0|W1|W2|W3]  (8192x128x128 f32 WMMA) -----------
__global__ void __launch_bounds__(128) k_wh(const float* __restrict__ hmat,
                                            const float* __restrict__ W,
                                            float* __restrict__ Wh, int n_nodes) {
  const int wave = threadIdx.x >> 5, lane = threadIdx.x & 31;
  const int m = lane & 15, half = lane >> 4;
  const int ntile_n = IN_DIM / 16;  // 8
  int tile = blockIdx.x * (blockDim.x >> 5) + wave;
  const int total = (n_nodes / 16) * ntile_n;
  if (tile >= total) return;
  const int mt = tile / ntile_n, nt = tile % ntile_n;
  const int n = nt * 16 + m;
  const int hh = n >> 5, cc = n & 31;
  const float* arow = hmat + (size_t)(mt * 16 + m) * IN_DIM;
  const float* wcol = W + (size_t)hh * IN_DIM * HD + cc;
  v8f acc = {};
  for (int k0 = 0; k0 < IN_DIM; k0 += 4) {
    const int kr = k0 + 2 * half;
    v2f a; a.x = arow[kr]; a.y = arow[kr + 1];
    v2f b; b.x = wcol[(size_t)kr * HD]; b.y = wcol[(size_t)(kr + 1) * HD];
    acc = __builtin_amdgcn_wmma_f32_16x16x4_f32(false, a, false, b, (short)0, acc, false, false);
  }
#pragma unroll
  for (int v = 0; v < 8; ++v)
    Wh[(size_t)(mt * 16 + v + 8 * half) * IN_DIM + n] = acc[v];
}

// ---------------- per-edge att + gate MLPs (WMMA), hash lookup --------------
__global__ void __launch_bounds__(128)
k_edgeA(const int* __restrict__ src, const int* __restrict__ tgt, const int* __restrict__ etype,
        const float* __restrict__ Wh, const float* __restrict__ rel_emb,
        const float* __restrict__ aw1, const float* __restrict__ ab1, const float* __restrict__ aw2,
        const float* __restrict__ gw1, const float* __restrict__ gb1, const float* __restrict__ gw2,
        const float* __restrict__ gb2, const unsigned* __restrict__ hkeys,
        const float* __restrict__ hvals, float* __restrict__ e_buf, float* __restrict__ gated_buf,
        int* __restrict__ n_matched, int n_edges, int n_nodes, int tiles_per_wave) {
  const int h = blockIdx.y;
  __shared__ float s_aw1[ATT_K * HD];   // 24KB, TDM-staged
  __shared__ float s_gw1[GATE_K * HD];  // 8KB,  TDM-staged
  __shared__ float s_aw2[HD], s_gw2[HD], s_ab1[HD], s_gb1[HD];
  __shared__ float s_z[4][2][16][HD + 1];  // per-wave staging of z tiles

  // Tensor Data Mover: DMA both first-layer weight matrices into LDS (wave 0).
  if (threadIdx.x < 32) {
    tdm_load_1d(aw1 + (size_t)h * ATT_K * HD, s_aw1, ATT_K * HD);
    tdm_load_1d(gw1 + (size_t)h * GATE_K * HD, s_gw1, GATE_K * HD);
  }
  if (threadIdx.x < HD) {
    s_aw2[threadIdx.x] = aw2[h * HD + threadIdx.x];
    s_gw2[threadIdx.x] = gw2[h * HD + threadIdx.x];
    s_ab1[threadIdx.x] = ab1[h * HD + threadIdx.x];
    s_gb1[threadIdx.x] = gb1[h * HD + threadIdx.x];
  }
  if (threadIdx.x < 32) __builtin_amdgcn_s_wait_tensorcnt(0);
  __syncthreads();
  const float gb2v = gb2[h];

  const int wave = threadIdx.x >> 5, lane = threadIdx.x & 31;
  const int m = lane & 15, half = lane >> 4;
  const int wave_id = blockIdx.x * (blockDim.x >> 5) + wave;
  const int n_waves = gridDim.x * (blockDim.x >> 5);

  for (int it = 0; it < tiles_per_wave; ++it) {
    const int tile = wave_id + it * n_waves;
    const int edge = tile * 16 + m;
    if (it + 1 < tiles_per_wave) {  // prefetch next tile's edge stream
      __builtin_prefetch(&src[edge + 16 * n_waves], 0, 1);
      __builtin_prefetch(&tgt[edge + 16 * n_waves], 0, 1);
    }
    const int sv = src[edge], tv = tgt[edge], ty = etype[edge];
    const float* pse = Wh + (size_t)sv * IN_DIM + h * HD;
    const float* pte = Wh + (size_t)tv * IN_DIM + h * HD;
    const float* pre = rel_emb + (size_t)ty * IN_DIM;

    // attention MLP first layer: 16x192 @ 192x32 -> 16x32
    v8f a0 = {}, a1 = {};
    for (int k0 = 0; k0 < ATT_K; k0 += 4) {
      const float* base; int off;
      if (k0 < HD)          { base = pse; off = k0; }
      else if (k0 < 2 * HD) { base = pte; off = k0 - HD; }
      else                  { base = pre; off = k0 - 2 * HD; }
      const int ka = off + 2 * half;
      v2f a; a.x = base[ka]; a.y = base[ka + 1];
      const int kr = k0 + 2 * half;
      v2f b0, b1;
      b0.x = s_aw1[kr * HD + m];        b0.y = s_aw1[(kr + 1) * HD + m];
      b1.x = s_aw1[kr * HD + 16 + m];   b1.y = s_aw1[(kr + 1) * HD + 16 + m];
      a0 = __builtin_amdgcn_wmma_f32_16x16x4_f32(false, a, false, b0, (short)0, a0, false, false);
      a1 = __builtin_amdgcn_wmma_f32_16x16x4_f32(false, a, false, b1, (short)0, a1, false, false);
    }
    // gate MLP first layer: 16x64 @ 64x32 -> 16x32
    v8f g0 = {}, g1 = {};
    for (int k0 = 0; k0 < GATE_K; k0 += 4) {
      const float* base = (k0 < HD) ? pse : pte;
      const int off = (k0 < HD) ? k0 : k0 - HD;
      const int ka = off + 2 * half;
      v2f a; a.x = base[ka]; a.y = base[ka + 1];
      const int kr = k0 + 2 * half;
      v2f b0, b1;
      b0.x = s_gw1[kr * HD + m];        b0.y = s_gw1[(kr + 1) * HD + m];
      b1.x = s_gw1[kr * HD + 16 + m];   b1.y = s_gw1[(kr + 1) * HD + 16 + m];
      g0 = __builtin_amdgcn_wmma_f32_16x16x4_f32(false, a, false, b0, (short)0, g0, false, false);
      g1 = __builtin_amdgcn_wmma_f32_16x16x4_f32(false, a, false, b1, (short)0, g1, false, false);
    }
    // stage C tiles row-major into LDS (bias + leaky fused)
#pragma unroll
    for (int v = 0; v < 8; ++v) {
      const int row = v + 8 * half;
      s_z[wave][0][row][m]      = leakyf(a0[v] + s_ab1[m]);
      s_z[wave][0][row][16 + m] = leakyf(a1[v] + s_ab1[16 + m]);
      s_z[wave][1][row][m]      = leakyf(g0[v] + s_gb1[m]);
      s_z[wave][1][row][16 + m] = leakyf(g1[v] + s_gb1[16 + m]);
    }
    __syncthreads();
    // half 0: e = z_att @ aw2 ; half 1: gate pre-act = z_gate @ gw2
    const float* wsel = (half == 0) ? s_aw2 : s_gw2;
    float dot = 0.f;
#pragma unroll
    for (int n = 0; n < HD; ++n) dot += s_z[wave][half][m][n] * wsel[n];
    const size_t gi = (size_t)h * n_edges + edge;
    if (half == 0) {
      e_buf[gi] = dot;
    } else {
      const float gate = sigmoidf(dot + gb2v);
      unsigned key = (unsigned)sv * (unsigned)n_nodes + (unsigned)tv;
      unsigned slot = (key * 2654435761u) & HMASK;
      float matched = 0.f;
      bool found = false;
      while (true) {
        const unsigned k = hkeys[slot];
        if (k == key) { matched = hvals[(size_t)h * HSZ + slot]; found = true; break; }
        if (k == HEMPTY) break;
        slot = (slot + 1) & HMASK;
      }
      gated_buf[gi] = gate * matched;
      if (found && h == 0) atomicAdd(n_matched, 1);
    }
    __syncthreads();
  }
}

// ---------------- pass B: finalize e, segment max ---------------------------
__global__ void k_edgeB(float* __restrict__ e_buf, const float* __restrict__ gated,
                        const float* __restrict__ lambda_g, const float* __restrict__ sum_ws,
                        const int* __restrict__ n_matched, const int* __restrict__ src,
                        float* __restrict__ mx, int n_edges, int n_nodes, float inv_nc) {
  int i = blockIdx.x * blockDim.x + threadIdx.x;
  if (i >= NH * n_edges) return;
  int h = i / n_edges, e = i - h * n_edges;
  float lg = sigmoidf(lambda_g[h]) * 1.9f + 0.1f;  // (scale-min)+min
  float ev = e_buf[i];
  if (*n_matched > 0) ev += lg * gated[i];
  else ev += lg * (sum_ws[h] * inv_nc) * 0.01f;
  e_buf[i] = ev;
  atomicMaxF(&mx[h * n_nodes + src[e]], ev);
}

// ---------------- pass C: exp, segment sum ----------------------------------
__global__ void k_edgeC(float* __restrict__ e_buf, const float* __restrict__ mx,
                        const int* __restrict__ src, float* __restrict__ sm,
                        int n_edges, int n_nodes) {
  int i = blockIdx.x * blockDim.x + threadIdx.x;
  if (i >= NH * n_edges) return;
  int h = i / n_edges, e = i - h * n_edges;
  float d = e_buf[i] - mx[h * n_nodes + src[e]];
  float ex = __expf(fminf(d, 20.f));
  e_buf[i] = ex;
  atomicAdd(&sm[h * n_nodes + src[e]], ex);
}

// ---------------- pass D: side scatter --------------------------------------
__global__ void k_edgeD(const float* __restrict__ ex_buf, const float* __restrict__ sm,
                        const int* __restrict__ src, const int* __restrict__ tgt,
                        const float* __restrict__ Wh, float* __restrict__ side,
                        int n_edges, int n_nodes) {
  int i = blockIdx.x * blockDim.x + threadIdx.x;
  if (i >= NH * n_edges) return;
  int h = i / n_edges, e = i - h * n_edges;
  int s = src[e], t = tgt[e];
  float attn = ex_buf[i] / (sm[h * n_nodes + s] + 1e-10f);
  const float* te = Wh + (size_t)t * IN_DIM + h * HD;
  float* sd = side + ((size_t)h * n_nodes + s) * HD;
#pragma unroll
  for (int c = 0; c < HD; ++c) atomicAdd(&sd[c], attn * te[c]);
}

// ---------------- node epilogue ---------------------------------------------
__global__ void __launch_bounds__(128) k_node_out(const float* __restrict__ Wh,
                                                  const float* __restrict__ side,
                                                  const float* __restrict__ W_bi,
                                                  float* __restrict__ out, int n_nodes) {
  const int node = blockIdx.x;
  const int j = threadIdx.x;  // 0..127
  const int hh = j >> 5, c = j & 31;
  __shared__ float s_wh[IN_DIM], s_sd[IN_DIM];
  s_wh[j] = Wh[(size_t)node * IN_DIM + j];
  s_sd[j] = side[((size_t)hh * n_nodes + node) * HD + c];
  __syncthreads();
  const float* wb = W_bi + (size_t)hh * HD * HD;
  float y = 0.f;
#pragma unroll
  for (int k = 0; k < HD; ++k) y += s_wh[hh * HD + k] * s_sd[hh * HD + k] * wb[k * HD + c];
  float whv = s_wh[j], sd = s_sd[j];
  float o = leakyf(whv + sd) + leakyf(y) + whv;
  out[(size_t)node * IN_DIM + j] = o > 0.f ? o : (__expf(o) - 1.f);
}

// ---------------- launcher ---------------------------------------------------
extern "C" void kernel_launch(void* const* d_in, const int* in_sizes, int n_in,
                              void* d_out, int out_size, void* d_ws, size_t ws_size,
                              hipStream_t stream) {
  const float* h        = (const float*)d_in[0];
  const float* W        = (const float*)d_in[1];
  const float* W_bi     = (const float*)d_in[2];
  const float* aw1      = (const float*)d_in[3];
  const float* ab1      = (const float*)d_in[4];
  const float* aw2      = (const float*)d_in[5];
  const float* lambda_c = (const float*)d_in[6];
  const float* lambda_g = (const float*)d_in[7];
  const float* gw1      = (const float*)d_in[8];
  const float* gb1      = (const float*)d_in[9];
  const float* gw2      = (const float*)d_in[10];
  const float* gb2      = (const float*)d_in[11];
  const float* rel_emb  = (const float*)d_in[12];
  const int* edge_index = (const int*)d_in[13];
  const int* edge_type  = (const int*)d_in[14];
  const int* c_idx      = (const int*)d_in[15];
  const float* c_val    = (const float*)d_in[16];
  const int* c_typ      = (const int*)d_in[17];

  const int n_edges = in_sizes[14];
  const int n_c     = in_sizes[16];
  const int n_nodes = in_sizes[0] / IN_DIM;
  const int n_ctype = in_sizes[6] / NH;
  const int* src = edge_index;
  const int* tgt = edge_index + n_edges;
  const int* c_src = c_idx;
  const int* c_tgt = c_idx + n_c;

  char* w = (char*)d_ws;
  float*    Wh      = (float*)w;    w += (size_t)n_nodes * IN_DIM * 4;
  unsigned* hkeys   = (unsigned*)w; w += (size_t)HSZ * 4;
  float*    hvals   = (float*)w;    w += (size_t)NH * HSZ * 4;
  float*    e_buf   = (float*)w;    w += (size_t)NH * n_edges * 4;
  float*    gated   = (float*)w;    w += (size_t)NH * n_edges * 4;
  float*    mx      = (float*)w;    w += (size_t)NH * n_nodes * 4;
  float*    sm      = (float*)w;    w += (size_t)NH * n_nodes * 4;
  float*    side    = (float*)w;    w += (size_t)NH * n_nodes * HD * 4;
  float*    sum_ws  = (float*)w;    w += NH * 4;
  int*      n_match = (int*)w;      w += 16;

  (void)hipMemsetAsync(hkeys, 0xFF, (size_t)HSZ * 4, stream);
  (void)hipMemsetAsync(hvals, 0, (size_t)NH * HSZ * 4, stream);
  (void)hipMemsetAsync(sm, 0, (size_t)NH * n_nodes * 4, stream);
  (void)hipMemsetAsync(side, 0, (size_t)NH * n_nodes * HD * 4, stream);
  (void)hipMemsetAsync(sum_ws, 0, NH * 4 + 16, stream);

  k_init_mx<<<(NH * n_nodes + 255) / 256, 256, 0, stream>>>(mx, NH * n_nodes);
  k_hash_build<<<(n_c + 255) / 256, 256, 0, stream>>>(c_src, c_tgt, c_val, c_typ, lambda_c,
                                                      hkeys, hvals, sum_ws, n_c, n_nodes, n_ctype);

  const int wh_tiles = (n_nodes / 16) * (IN_DIM / 16);
  k_wh<<<(wh_tiles + 3) / 4, 128, 0, stream>>>(h, W, Wh, n_nodes);

  const int blocksA = 512;
  const int n_waves = blocksA * 4;
  const int n_tiles = n_edges / 16;
  const int tpw = n_tiles / n_waves;  // 524288/16/2048 = 16
  dim3 gA(blocksA, NH);
  k_edgeA<<<gA, 128, 0, stream>>>(src, tgt, edge_type, Wh, rel_emb, aw1, ab1, aw2,
                                  gw1, gb1, gw2, gb2, hkeys, hvals, e_buf, gated, n_match,
                                  n_edges, n_nodes, tpw);

  const int totE = NH * n_edges;
  k_edgeB<<<(totE + 255) / 256, 256, 0, stream>>>(e_buf, gated, lambda_g, sum_ws, n_match,
                                                  src, mx, n_edges, n_nodes, 1.0f / (float)n_c);
  k_edgeC<<<(totE + 255) / 256, 256, 0, stream>>>(e_buf, mx, src, sm, n_edges, n_nodes);
  k_edgeD<<<(totE + 255) / 256, 256, 0, stream>>>(e_buf, sm, src, tgt, Wh, side, n_edges, n_nodes);

  k_node_out<<<n_nodes, 128, 0, stream>>>(Wh, side, W_bi, (float*)d_out, n_nodes);
}